// OrthonormalizeLayer_29875792511017
// MI455X (gfx1250) — compile-verified
//
#include <hip/hip_runtime.h>
#include <math.h>

// CDNA5 / gfx1250, wave32. f32 WMMA fragments.
typedef float v2f __attribute__((ext_vector_type(2)));
typedef float v8f __attribute__((ext_vector_type(8)));

#define M_ROWS 768
#define R_COLS 64
#define PANEL 16
#define NPANEL (R_COLS / PANEL)
#define PITCH 17                      // 768x16 panel, padded row pitch -> no LDS bank conflicts
#define NTHREADS 256                  // 8 wave32 waves
#define NWAVES 8
#define ROWS_PER_WAVE (M_ROWS / NWAVES)  // 96 rows per wave (24 K-chunks of 4 / 6 blocks of 16)
#define EPS_NRM 1e-6f

__device__ __forceinline__ v8f wmma4(v2f a, v2f b, v8f c) {
    // D(16x16,f32) = A(16x4,f32) x B(4x16,f32) + C
    return __builtin_amdgcn_wmma_f32_16x16x4_f32(false, a, false, b, (short)0, c, false, false);
}

__global__ __launch_bounds__(NTHREADS)
void gram_schmidt_cdna5(const float* __restrict__ xr_g,
                        const float* __restrict__ xi_g,
                        float* out) {
    // Planes: X = original panel (columns overwritten with finished Q as we go),
    //         V = running (projection-subtracted) panel.
    __shared__ float Xr[M_ROWS * PITCH];
    __shared__ float Xi[M_ROWS * PITCH];
    __shared__ float Vr[M_ROWS * PITCH];
    __shared__ float Vi[M_ROWS * PITCH];
    __shared__ float Cbuf[PANEL * PANEL * 2];   // 16x16 complex coefficient tile
    __shared__ float qn_s[R_COLS];              // squared norm of each stored Q column
    __shared__ float red[34];                   // 16 complex dots + |v|^2 accumulator

    const int b    = blockIdx.x;
    const int tid  = threadIdx.x;
    const int lane = tid & 31;
    const int wave = tid >> 5;
    const int jj   = lane & 15;   // N index / row-in-16 index of fragment layouts
    const int hi   = lane >> 4;   // lane-half select (M/K +offset per ISA 7.12.2)

    const size_t gbase = (size_t)b * M_ROWS * R_COLS;   // element (b,0,0)

    for (int p = 0; p < NPANEL; ++p) {
        const int c0 = p * PANEL;

        // ---------------- load panel: X (original) and V (running copy) ----------------
        for (int idx = tid; idx < M_ROWS * PANEL; idx += NTHREADS) {
            int m = idx >> 4, c = idx & 15;
            float re = xr_g[gbase + (size_t)m * R_COLS + c0 + c];
            float im = xi_g[gbase + (size_t)m * R_COLS + c0 + c];
            Xr[m * PITCH + c] = re;  Xi[m * PITCH + c] = im;
            Vr[m * PITCH + c] = re;  Vi[m * PITCH + c] = im;
        }
        __syncthreads();

        // ---------- cross-panel projections via f32 WMMA (classical GS, blocked) ----------
        for (int q = 0; q < p; ++q) {
            Cbuf[tid] = 0.0f;            // zero 512-float coefficient tile
            Cbuf[tid + NTHREADS] = 0.0f;
            __syncthreads();

            // Split-K over M: wave accumulates partial C = Q_q^H * X over its 96 rows.
            v8f cr = {0.f,0.f,0.f,0.f,0.f,0.f,0.f,0.f};
            v8f ci = {0.f,0.f,0.f,0.f,0.f,0.f,0.f,0.f};
            for (int kc = 0; kc < ROWS_PER_WAVE / 4; ++kc) {
                int m0 = wave * ROWS_PER_WAVE + kc * 4;
                // A[j,k] = Q[m0+k, q*16+j]   (conj via sign below)
                int rA0 = m0 + hi * 2 + 0;
                int rA1 = m0 + hi * 2 + 1;
                size_t qa0 = (gbase + (size_t)rA0 * R_COLS + q * PANEL + jj) * 2;
                size_t qa1 = (gbase + (size_t)rA1 * R_COLS + q * PANEL + jj) * 2;
                v2f ar = { out[qa0],     out[qa1]     };
                v2f ai = { out[qa0 + 1], out[qa1 + 1] };
                // B[k,n] = X[m0+k, n]
                int rB0 = m0 + hi;
                int rB1 = m0 + 2 + hi;
                v2f br = { Xr[rB0 * PITCH + jj], Xr[rB1 * PITCH + jj] };
                v2f bi = { Xi[rB0 * PITCH + jj], Xi[rB1 * PITCH + jj] };
                v2f nai = { -ai.x, -ai.y };
                cr = wmma4(ar,  br, cr);   // Re(Q^H X) = Qr^T Xr + Qi^T Xi
                cr = wmma4(ai,  bi, cr);
                ci = wmma4(ar,  bi, ci);   // Im(Q^H X) = Qr^T Xi - Qi^T Xr
                ci = wmma4(nai, br, ci);
            }
            // reduce partials across the 8 waves (C/D layout: row = v + hi*8, col = jj)
            for (int v = 0; v < 8; ++v) {
                int j = v + hi * 8;
                atomicAdd(&Cbuf[(j * PANEL + jj) * 2 + 0], cr[v]);
                atomicAdd(&Cbuf[(j * PANEL + jj) * 2 + 1], ci[v]);
            }
            __syncthreads();

            // coef[j,i] = dots / ||Q_j||^2, masked like the reference
            {
                int j = tid >> 4, i = tid & 15;
                float q2 = qn_s[q * PANEL + j];
                float s = (q2 > 0.0f) ? (1.0f / q2) : 0.0f;
                Cbuf[(j * PANEL + i) * 2 + 0] *= s;
                Cbuf[(j * PANEL + i) * 2 + 1] *= s;
            }
            __syncthreads();

            // V -= Q_q * coef   (wave: 6 row-blocks of 16, K=16 in 4 WMMA chunks)
            for (int blk = 0; blk < ROWS_PER_WAVE / PANEL; ++blk) {
                int m0 = wave * ROWS_PER_WAVE + blk * PANEL;
                v8f dr = {0.f,0.f,0.f,0.f,0.f,0.f,0.f,0.f};
                v8f di = {0.f,0.f,0.f,0.f,0.f,0.f,0.f,0.f};
                for (int kc = 0; kc < 4; ++kc) {
                    int mg = m0 + jj;
                    int k0 = kc * 4 + hi * 2;                 // A[m,k]: two consecutive k
                    size_t qa = (gbase + (size_t)mg * R_COLS + q * PANEL + k0) * 2;
                    v2f ar = { out[qa + 0], out[qa + 2] };
                    v2f ai = { out[qa + 1], out[qa + 3] };
                    int kb0 = kc * 4 + hi;                    // B[k,n] = coef[k,n]
                    int kb1 = kc * 4 + 2 + hi;
                    v2f br = { Cbuf[(kb0 * PANEL + jj) * 2 + 0], Cbuf[(kb1 * PANEL + jj) * 2 + 0] };
                    v2f bi = { Cbuf[(kb0 * PANEL + jj) * 2 + 1], Cbuf[(kb1 * PANEL + jj) * 2 + 1] };
                    v2f nai = { -ai.x, -ai.y };
                    dr = wmma4(ar,  br, dr);   // Re(Q C) = Qr Cr - Qi Ci
                    dr = wmma4(nai, bi, dr);
                    di = wmma4(ar,  bi, di);   // Im(Q C) = Qr Ci + Qi Cr
                    di = wmma4(ai,  br, di);
                }
                for (int v = 0; v < 8; ++v) {  // unique (row,col) per lane -> plain RMW
                    int row = m0 + v + hi * 8;
                    Vr[row * PITCH + jj] -= dr[v];
                    Vi[row * PITCH + jj] -= di[v];
                }
            }
            __syncthreads();
        }

        // ---------------- in-panel sequential classical GS (16 steps) ----------------
        for (int i = 0; i < PANEL; ++i) {
            if (tid < 34) red[tid] = 0.0f;
            __syncthreads();

            // dots_j = Q_j^H x_i  (x_i = ORIGINAL column, Q_j = finished columns in X plane)
            for (int j = 0; j < i; ++j) {
                float sre = 0.0f, sim = 0.0f;
                for (int r = 0; r < 3; ++r) {
                    int m = tid + r * NTHREADS;
                    float xr = Xr[m * PITCH + i], xi = Xi[m * PITCH + i];
                    float qr = Xr[m * PITCH + j], qi = Xi[m * PITCH + j];
                    sre += qr * xr + qi * xi;
                    sim += qr * xi - qi * xr;
                }
                atomicAdd(&red[2 * j + 0], sre);
                atomicAdd(&red[2 * j + 1], sim);
            }
            __syncthreads();

            float vr[3], vi[3];
            for (int r = 0; r < 3; ++r) {
                int m = tid + r * NTHREADS;
                vr[r] = Vr[m * PITCH + i];
                vi[r] = Vi[m * PITCH + i];
            }
            for (int j = 0; j < i; ++j) {
                float q2 = qn_s[c0 + j];
                float s = (q2 > 0.0f) ? (1.0f / q2) : 0.0f;
                float crj = red[2 * j + 0] * s;
                float cij = red[2 * j + 1] * s;
                for (int r = 0; r < 3; ++r) {
                    int m = tid + r * NTHREADS;
                    float qr = Xr[m * PITCH + j], qi = Xi[m * PITCH + j];
                    vr[r] -= crj * qr - cij * qi;
                    vi[r] -= crj * qi + cij * qr;
                }
            }
            float n2 = 0.0f;
            for (int r = 0; r < 3; ++r) n2 += vr[r] * vr[r] + vi[r] * vi[r];
            atomicAdd(&red[32], n2);
            __syncthreads();

            float nrm2 = red[32];
            float nrm  = sqrtf(nrm2);
            float inv  = (nrm > EPS_NRM) ? (1.0f / fmaxf(nrm, 1e-30f)) : 1.0f;
            for (int r = 0; r < 3; ++r) {
                int m = tid + r * NTHREADS;
                float qr = vr[r] * inv, qi = vi[r] * inv;
                Xr[m * PITCH + i] = qr;          // column i now holds finished Q_i
                Xi[m * PITCH + i] = qi;
                size_t o = (gbase + (size_t)m * R_COLS + c0 + i) * 2;
                out[o + 0] = qr;                 // (B,M,R,2) == interleaved complex
                out[o + 1] = qi;
            }
            if (tid == 0) qn_s[c0 + i] = nrm2 * inv * inv;
            __syncthreads();
        }

        // make this panel's global Q writes visible to next panel's WMMA gathers
        __threadfence();
        __syncthreads();
    }
}

extern "C" void kernel_launch(void* const* d_in, const int* in_sizes, int n_in,
                              void* d_out, int out_size, void* d_ws, size_t ws_size,
                              hipStream_t stream) {
    (void)n_in; (void)out_size; (void)d_ws; (void)ws_size;
    const float* xr = (const float*)d_in[0];   // x_real (B, M, R) f32
    const float* xi = (const float*)d_in[1];   // x_imag (B, M, R) f32
    float* out = (float*)d_out;                // (B, M, R, 2) f32
    int B = in_sizes[0] / (M_ROWS * R_COLS);
    gram_schmidt_cdna5<<<dim3(B), dim3(NTHREADS), 0, stream>>>(xr, xi, out);
}